// L2ChamferDist_24790551233440
// MI455X (gfx1250) — compile-verified
//
#include <hip/hip_runtime.h>
#include <math.h>

typedef __attribute__((ext_vector_type(2))) float v2f;
typedef __attribute__((ext_vector_type(8))) float v8f;

#define B_SZ   16
#define N_PTS  4096
#define K_PTS  4096
#define CD_WT  0.2f
#define EPS_F  1e-7f

// ds_swizzle group-of-32: offset = {xor[14:10], or[9:5], and[4:0]}, offset[15]=0
template <int XOR>
__device__ __forceinline__ float swz_xor(float v) {
  return __int_as_float(
      __builtin_amdgcn_ds_swizzle(__float_as_int(v), (XOR << 10) | 0x1f));
}

// ---------------------------------------------------------------------------
// Chamfer: one block = 128 adv points of one batch (8 waves x 16-pt tiles).
// Block stages all 4096 ori points as [x,y,z,|o|^2] float4 in LDS (64KB),
// each wave sweeps 256 ori tiles with V_WMMA_F32_16X16X4_F32.
// D[M,N] = ox*(-2ax) + oy*(-2ay) + oz*(-2az) + |o|^2*1 = |o|^2 - 2<a,o>
// min_k sqdist(n) = a2(n) + min_M D ; k-reduction is in-lane over 8 acc VGPRs
// plus one SWAPX16 swizzle across lane halves.
// ---------------------------------------------------------------------------
__global__ __launch_bounds__(256) void chamfer_kernel(
    const float* __restrict__ adv, const float* __restrict__ ori,
    const float* __restrict__ w, float* __restrict__ ws) {
  __shared__ float4 lds_ori[K_PTS];   // 64 KB
  __shared__ float wave_part[8];

  const int b    = blockIdx.x >> 5;   // 32 blocks per batch
  const int chunk = blockIdx.x & 31;
  const int tid  = threadIdx.x;
  const int lane = tid & 31;
  const int wave = tid >> 5;
  const int M    = lane & 15;
  const int half = lane >> 4;

  // Stage ori points + squared norm into LDS.
  const float* orib = ori + (size_t)b * K_PTS * 3;
  for (int k = tid; k < K_PTS; k += 256) {
    float x = orib[3 * k + 0];
    float y = orib[3 * k + 1];
    float z = orib[3 * k + 2];
    lds_ori[k] = make_float4(x, y, z, x * x + y * y + z * z);
  }
  __syncthreads();

  // B fragment: column N = adv point (lane%16). Rows K: [-2x,-2y,-2z,1].
  const int n = (chunk * 8 + wave) * 16 + M;
  const float* advb = adv + ((size_t)b * N_PTS + n) * 3;
  float ax = advb[0], ay = advb[1], az = advb[2];
  float a2 = ax * ax + ay * ay + az * az;
  v2f bfrag;
  bfrag.x = half ? (-2.0f * az) : (-2.0f * ax);  // K=2 : K=0
  bfrag.y = half ? 1.0f         : (-2.0f * ay);  // K=3 : K=1

  v8f rmin;
#pragma unroll
  for (int i = 0; i < 8; ++i) rmin[i] = 3.4e38f;

  const float2* lds2 = (const float2*)lds_ori;
#pragma unroll 4
  for (int kt = 0; kt < K_PTS / 16; ++kt) {
    // A fragment: lanes0-15 get (x,y) = K0,K1 ; lanes16-31 get (z,o2) = K2,K3
    float2 av = lds2[(((kt << 4) + M) << 1) + half];  // ds_load_b64
    v2f afrag;
    afrag.x = av.x;
    afrag.y = av.y;
    v8f c = {};
    v8f d = __builtin_amdgcn_wmma_f32_16x16x4_f32(
        /*neg_a=*/false, afrag, /*neg_b=*/false, bfrag,
        /*c_mod=*/(short)0, c, /*reuse_a=*/false, /*reuse_b=*/false);
#pragma unroll
    for (int i = 0; i < 8; ++i) rmin[i] = fminf(rmin[i], d[i]);
  }

  // min over M = min over 8 VGPRs (in-lane) then across the lane halves.
  float s = rmin[0];
#pragma unroll
  for (int i = 1; i < 8; ++i) s = fminf(s, rmin[i]);
  s = fminf(s, swz_xor<16>(s));      // combine M=0..7 with M=8..15

  // per-adv-point contribution; each point is duplicated in both halves
  float t = s + a2;
  t += swz_xor<1>(t);
  t += swz_xor<2>(t);
  t += swz_xor<4>(t);
  t += swz_xor<8>(t);
  t += swz_xor<16>(t);               // t = 2 * sum over the wave's 16 points

  if (lane == 0) wave_part[wave] = t * 0.5f;
  __syncthreads();
  if (tid == 0) {
    float bs = 0.0f;
#pragma unroll
    for (int i = 0; i < 8; ++i) bs += wave_part[i];
    ws[16 + blockIdx.x] = bs * w[b] * (CD_WT / (float)(B_SZ * N_PTS));
  }
}

// ---------------------------------------------------------------------------
// L2 term: one block per batch, sqrt(sum((adv-ori)^2) + eps) * w[b] / B
// ---------------------------------------------------------------------------
__global__ __launch_bounds__(256) void l2_kernel(
    const float* __restrict__ adv_obj, const float* __restrict__ ori_obj,
    const float* __restrict__ w, float* __restrict__ ws) {
  __shared__ float red[256];
  const int b = blockIdx.x;
  const int tid = threadIdx.x;
  const int per_b = 8 * 512 * 3;  // 12288
  const float* a = adv_obj + (size_t)b * per_b;
  const float* o = ori_obj + (size_t)b * per_b;
  float s = 0.0f;
  for (int i = tid; i < per_b; i += 256) {
    float d = a[i] - o[i];
    s += d * d;
  }
  red[tid] = s;
  __syncthreads();
  for (int st = 128; st > 0; st >>= 1) {
    if (tid < st) red[tid] += red[tid + st];
    __syncthreads();
  }
  if (tid == 0) ws[b] = sqrtf(red[0] + EPS_F) * w[b] * (1.0f / (float)B_SZ);
}

// ---------------------------------------------------------------------------
// Final deterministic reduction of 16 l2 partials + 512 chamfer partials.
// ---------------------------------------------------------------------------
__global__ __launch_bounds__(256) void final_kernel(const float* __restrict__ ws,
                                                    float* __restrict__ out) {
  __shared__ float red[256];
  const int tid = threadIdx.x;
  float s = 0.0f;
  for (int i = tid; i < 16 + 512; i += 256) s += ws[i];
  red[tid] = s;
  __syncthreads();
  for (int st = 128; st > 0; st >>= 1) {
    if (tid < st) red[tid] += red[tid + st];
    __syncthreads();
  }
  if (tid == 0) out[0] = red[0];
}

extern "C" void kernel_launch(void* const* d_in, const int* in_sizes, int n_in,
                              void* d_out, int out_size, void* d_ws,
                              size_t ws_size, hipStream_t stream) {
  const float* adv_pc  = (const float*)d_in[0];
  const float* ori_pc  = (const float*)d_in[1];
  const float* adv_obj = (const float*)d_in[2];
  const float* ori_obj = (const float*)d_in[3];
  const float* weights = (const float*)d_in[4];
  float* ws  = (float*)d_ws;
  float* out = (float*)d_out;

  l2_kernel<<<B_SZ, 256, 0, stream>>>(adv_obj, ori_obj, weights, ws);
  chamfer_kernel<<<B_SZ * (N_PTS / 128), 256, 0, stream>>>(adv_pc, ori_pc,
                                                           weights, ws);
  final_kernel<<<1, 256, 0, stream>>>(ws, out);
}